// Gaussian3d_420906795754
// MI455X (gfx1250) — compile-verified
//
#include <hip/hip_runtime.h>
#include <stdint.h>

namespace {
constexpr int kD = 64;
constexpr int kH = 128;
constexpr int kW = 128;
constexpr int KS = 5;

constexpr int TW = 32;          // tile width  (W)
constexpr int TH = 32;          // tile height (H)
constexpr int DSPLIT = 2;       // split depth sweep for more blocks
constexpr int DCHUNK = kD / DSPLIT;

// W halo widened to +-4 so every row is ten 16B-aligned vec4 chunks.
constexpr int RAW_W = TW + 8;   // 40 floats per row
constexpr int RAW_H = TH + 4;   // 36 rows, halo +-2 in H
constexpr int RAW_N = RAW_W * RAW_H;     // 1440
constexpr int VEC_PER_ROW = RAW_W / 4;   // 10
constexpr int NVEC = RAW_H * VEC_PER_ROW; // 360 b128 loads per slice
constexpr int NT = 256;         // 8 wave32
}

// CDNA5 async global->LDS copies (no VGPR round trip, tracked by ASYNCcnt).
// vdst = per-lane LDS byte address, vaddr = 32-bit byte offset, saddr = base.
__device__ __forceinline__ void async_load_b128(uint32_t lds_off, uint32_t voff,
                                                uint64_t sbase) {
  asm volatile("global_load_async_to_lds_b128 %0, %1, %2 offset:0"
               :: "v"(lds_off), "v"(voff), "s"(sbase)
               : "memory");
}

__device__ __forceinline__ void wait_async0() {
  asm volatile("s_wait_asynccnt 0x0" ::: "memory");
}

__launch_bounds__(NT)
__global__ void gauss3d_sep_kernel(const float* __restrict__ x,
                                   const float* __restrict__ w3,
                                   float* __restrict__ out) {
  __shared__ float raw[2][RAW_N];      // double-buffered halo slice (36x40)
  __shared__ float s1[RAW_H * TW];     // after W-conv (36x32)
  __shared__ float gsh[KS];

  const int tid = threadIdx.x;
  const int tx = tid & 31;             // w within tile (lane)
  const int ty = tid >> 5;             // 0..7 (wave id)

  const int w0 = blockIdx.x * TW;
  const int h0 = blockIdx.y * TH;
  const int chunk = blockIdx.z % DSPLIT;
  const int nc = blockIdx.z / DSPLIT;  // n*C + c, contiguous D*H*W slab
  const int d0 = chunk * DCHUNK;

  const float* xin = x + (size_t)nc * kD * kH * kW;
  float* o = out + (size_t)nc * kD * kH * kW;

  // Recover the exact separable 1D kernel: g[i] = sum_{j,k} w[0,0,i,j,k].
  if (tid < KS) {
    float s = 0.f;
    #pragma unroll
    for (int t = 0; t < KS * KS; ++t) s += w3[tid * KS * KS + t];
    gsh[tid] = s;
  }
  // Zero both raw buffers: out-of-volume chunks are never rewritten,
  // giving SAME (zero) padding in H/W for free across all slices.
  for (int i = tid; i < 2 * RAW_N; i += NT) (&raw[0][0])[i] = 0.f;
  __syncthreads();

  const float g0 = gsh[0], g1 = gsh[1], g2 = gsh[2], g3 = gsh[3], g4 = gsh[4];

  auto issue_load = [&](int ds, int buf) {
    const uint64_t sbase = (uint64_t)(uintptr_t)(xin + (size_t)ds * kH * kW);
    for (int i = tid; i < NVEC; i += NT) {
      const int r = i / VEC_PER_ROW;          // 0..35
      const int cv = i - r * VEC_PER_ROW;     // 0..9
      const int gh = h0 - 2 + r;
      const int gwb = w0 - 4 + (cv << 2);     // multiple of 4 -> 16B aligned
      // each vec4 is entirely in-bounds or entirely out-of-bounds in W
      if ((unsigned)gh < (unsigned)kH && (unsigned)gwb < (unsigned)kW) {
        const uint32_t voff = (uint32_t)((gh * kW + gwb) * 4);
        // low 32 bits of a flat shared pointer are the LDS byte offset
        const uint32_t loff = (uint32_t)(uintptr_t)(&raw[buf][r * RAW_W + (cv << 2)]);
        async_load_b128(loff, voff, sbase);
      }
    }
  };

  // Preload the first slice of the pipeline (ds = d0-2 may be out of range).
  if (d0 - 2 >= 0) issue_load(d0 - 2, 0);
  wait_async0();
  __syncthreads();

  // Register ring for the D-direction conv: s2 of the last 4 slices.
  float ring[4][4];
  #pragma unroll
  for (int q = 0; q < 4; ++q)
    #pragma unroll
    for (int j = 0; j < 4; ++j) ring[q][j] = 0.f;

  for (int step = 0; step < DCHUNK + 4; ++step) {
    const int ds = d0 - 2 + step;      // slice being 2D-convolved this step
    const int cur = step & 1;

    // Prefetch next slice into the other buffer; overlaps this step's compute.
    const int dsn = ds + 1;
    if (step + 1 < DCHUNK + 4 && dsn >= 0 && dsn < kD) issue_load(dsn, cur ^ 1);

    const bool valid = (ds >= 0) && (ds < kD);   // block-uniform

    if (valid) {
      // W-conv: raw[cur] (36x40) -> s1 (36x32); lanes read/write consecutive
      // addresses -> bank-conflict free.
      for (int i = tid; i < RAW_H * TW; i += NT) {
        const int r = i >> 5;
        const int c = i & 31;
        // output w = w0+c needs gw = w0+c-2..w0+c+2 -> columns c+2 .. c+6
        const float* rr = &raw[cur][r * RAW_W + c + 2];
        s1[i] = g0 * rr[0] + g1 * rr[1] + g2 * rr[2] + g3 * rr[3] + g4 * rr[4];
      }
    }
    __syncthreads();

    float s2v[4];
    if (valid) {
      // H-conv: s1 -> registers, 4 rows per thread (h = ty + 8q).
      #pragma unroll
      for (int q = 0; q < 4; ++q) {
        const int h = ty + (q << 3);
        s2v[q] = g0 * s1[(h + 0) * TW + tx] + g1 * s1[(h + 1) * TW + tx] +
                 g2 * s1[(h + 2) * TW + tx] + g3 * s1[(h + 3) * TW + tx] +
                 g4 * s1[(h + 4) * TW + tx];
      }
    } else {
      #pragma unroll
      for (int q = 0; q < 4; ++q) s2v[q] = 0.f;  // SAME padding in D
    }

    // D-conv via register ring; emit output d = ds - 2 once ring is full.
    #pragma unroll
    for (int q = 0; q < 4; ++q) {
      if (step >= 4) {
        const float ov = g0 * ring[q][0] + g1 * ring[q][1] + g2 * ring[q][2] +
                         g3 * ring[q][3] + g4 * s2v[q];
        const int d = ds - 2;
        const int hh = h0 + ty + (q << 3);
        o[((size_t)d * kH + hh) * kW + (w0 + tx)] = ov;  // lanes coalesce in W
      }
      ring[q][0] = ring[q][1];
      ring[q][1] = ring[q][2];
      ring[q][2] = ring[q][3];
      ring[q][3] = s2v[q];
    }

    // Fence: async fill of the next buffer landed; all LDS reads retired.
    wait_async0();
    __syncthreads();
  }
}

extern "C" void kernel_launch(void* const* d_in, const int* in_sizes, int n_in,
                              void* d_out, int out_size, void* d_ws, size_t ws_size,
                              hipStream_t stream) {
  (void)in_sizes; (void)n_in; (void)out_size; (void)d_ws; (void)ws_size;
  const float* x = (const float*)d_in[0];
  const float* w = (const float*)d_in[1];
  float* out = (float*)d_out;

  dim3 grid(kW / TW, kH / TH, 2 * 16 * DSPLIT);  // 4 x 4 x 64 = 1024 blocks
  gauss3d_sep_kernel<<<grid, NT, 0, stream>>>(x, w, out);
}